// MessagePassingLayer_85993835200698
// MI455X (gfx1250) — compile-verified
//
#include <hip/hip_runtime.h>

typedef float v2f __attribute__((ext_vector_type(2)));
typedef float v8f __attribute__((ext_vector_type(8)));

#define N_NODES_C 50000
#define N_EDGES_C 600000
#define NODE_IN_C 128
#define EDGE_IN_C 32
#define OUT_DIM_C 128
#define NT 8              // 128/16 output n-tiles
#define LDS_STRIDE 132    // padded row stride (floats) -> conflict-free transpose

// ws layout (floats): swizzled B-fragment weights
// Sm1 [8][72][32][2] @ 0        (288x128)
// Sm2 [8][32][32][2] @ 36864    (128x128)
// Su1 [8][64][32][2] @ 53248    (256x128)
// Su2 [8][32][32][2] @ 86016    (128x128)   total 102400 floats
#define OFF_SM1 0
#define OFF_SM2 36864
#define OFF_SU1 53248
#define OFF_SU2 86016
#define WS_TOTAL 102400

__device__ __forceinline__ v8f wmma_f32(v2f a, v2f b, v8f c) {
  // D = A(16x4) * B(4x16) + C, fp32 end to end
  return __builtin_amdgcn_wmma_f32_16x16x4_f32(false, a, false, b, (short)0, c,
                                               false, false);
}

__global__ void zero_out_kernel(float4* __restrict__ p, int n4) {
  int t = blockIdx.x * blockDim.x + threadIdx.x;
  if (t < n4) p[t] = make_float4(0.f, 0.f, 0.f, 0.f);
}

// Swizzle W[K][128] row-major -> S[nt][kk][lane][2] so a wave's B fragment is
// one contiguous 256B load:  S[...] = W[4*kk + 2*(lane>>4) + c][nt*16 + (lane&15)]
__global__ void prep_weights_kernel(const float* __restrict__ Wm1,
                                    const float* __restrict__ Wm2,
                                    const float* __restrict__ Wu1,
                                    const float* __restrict__ Wu2,
                                    float* __restrict__ ws) {
  int t = blockIdx.x * blockDim.x + threadIdx.x;
  const float* W;
  int K, idx;
  if (t < OFF_SM2)         { W = Wm1; K = 288; idx = t - OFF_SM1; }
  else if (t < OFF_SU1)    { W = Wm2; K = 128; idx = t - OFF_SM2; }
  else if (t < OFF_SU2)    { W = Wu1; K = 256; idx = t - OFF_SU1; }
  else if (t < WS_TOTAL)   { W = Wu2; K = 128; idx = t - OFF_SU2; }
  else return;
  int c    = idx & 1;
  int lane = (idx >> 1) & 31;
  int rest = idx >> 6;          // nt*(K/4) + kk
  int kk   = rest % (K / 4);
  int nt   = rest / (K / 4);
  int k = 4 * kk + 2 * (lane >> 4) + c;
  int n = nt * 16 + (lane & 15);
  ws[t] = W[k * OUT_DIM_C + n];
}

// 4 waves/block, one 16-edge tile per wave, full 128-dim output per wave.
__global__ __launch_bounds__(128) void edge_mlp_kernel(
    const float* __restrict__ x, const long long* __restrict__ ei,
    const float* __restrict__ ea, const float* __restrict__ ws,
    const float* __restrict__ bm1, const float* __restrict__ bm2,
    float* __restrict__ aggr) {
  __shared__ float lds_h[4 * 16 * LDS_STRIDE];
  const int lane = threadIdx.x & 31;
  const int wv   = threadIdx.x >> 5;
  const int tile = blockIdx.x * 4 + wv;
  if (tile * 16 >= N_EDGES_C) return;   // uniform per wave: EXEC stays all-ones
  const int base = tile * 16;
  const int half = lane >> 4;
  const int ln   = lane & 15;

  const long long e  = base + ln;
  const long long ri = ei[e];
  const long long ci = ei[N_EDGES_C + e];
  const float* px_r = x + ri * NODE_IN_C;
  const float* px_c = x + ci * NODE_IN_C;
  const float* pea  = ea + e * EDGE_IN_C;
  const float* Sm1  = ws + OFF_SM1;
  const float* Sm2  = ws + OFF_SM2;

  // ---- layer 1: h = relu([x_r | x_c | ea] @ Wm1 + bm1), K = 288 ----
  v8f acc[NT];
#pragma unroll
  for (int nt = 0; nt < NT; ++nt) {
    float bv = bm1[nt * 16 + ln];
    acc[nt] = (v8f){bv, bv, bv, bv, bv, bv, bv, bv};
  }
  for (int kk = 0; kk < 32; ++kk) {                 // K 0..127 : x[row]
    v2f a = *(const v2f*)(px_r + 4 * kk + 2 * half);
#pragma unroll
    for (int nt = 0; nt < NT; ++nt) {
      v2f b = *(const v2f*)(Sm1 + ((nt * 72 + kk) * 32 + lane) * 2);
      acc[nt] = wmma_f32(a, b, acc[nt]);
    }
  }
  for (int kk = 32; kk < 64; ++kk) {                // K 128..255 : x[col]
    v2f a = *(const v2f*)(px_c + 4 * (kk - 32) + 2 * half);
#pragma unroll
    for (int nt = 0; nt < NT; ++nt) {
      v2f b = *(const v2f*)(Sm1 + ((nt * 72 + kk) * 32 + lane) * 2);
      acc[nt] = wmma_f32(a, b, acc[nt]);
    }
  }
  for (int kk = 64; kk < 72; ++kk) {                // K 256..287 : edge_attr
    v2f a = *(const v2f*)(pea + 4 * (kk - 64) + 2 * half);
#pragma unroll
    for (int nt = 0; nt < NT; ++nt) {
      v2f b = *(const v2f*)(Sm1 + ((nt * 72 + kk) * 32 + lane) * 2);
      acc[nt] = wmma_f32(a, b, acc[nt]);
    }
  }

  // ReLU + C-layout -> LDS (transpose staging for layer-2 A fragments)
  float* hbuf = lds_h + wv * (16 * LDS_STRIDE);
#pragma unroll
  for (int nt = 0; nt < NT; ++nt)
#pragma unroll
    for (int j = 0; j < 8; ++j) {
      float v = acc[nt][j];
      hbuf[(j + 8 * half) * LDS_STRIDE + nt * 16 + ln] = v > 0.f ? v : 0.f;
    }

  // ---- layer 2: m = h @ Wm2 + bm2, K = 128 ----
  v8f acc2[NT];
#pragma unroll
  for (int nt = 0; nt < NT; ++nt) {
    float bv = bm2[nt * 16 + ln];
    acc2[nt] = (v8f){bv, bv, bv, bv, bv, bv, bv, bv};
  }
  for (int kk = 0; kk < 32; ++kk) {
    v2f a = *(const v2f*)(hbuf + ln * LDS_STRIDE + 4 * kk + 2 * half);
#pragma unroll
    for (int nt = 0; nt < NT; ++nt) {
      v2f b = *(const v2f*)(Sm2 + ((nt * 32 + kk) * 32 + lane) * 2);
      acc2[nt] = wmma_f32(a, b, acc2[nt]);
    }
  }

  // scatter-add messages into aggr[col]
#pragma unroll
  for (int j = 0; j < 8; ++j) {
    long long cm = ei[N_EDGES_C + base + j + 8 * half];
    float* dst = aggr + cm * OUT_DIM_C;
#pragma unroll
    for (int nt = 0; nt < NT; ++nt)
      atomicAdd(dst + nt * 16 + ln, acc2[nt][j]);
  }
}

// 4 waves/block, one 16-node tile per wave. aggr lives in `out`; overwritten
// in place (each wave fully consumes its own rows before storing).
__global__ __launch_bounds__(128) void node_mlp_kernel(
    const float* __restrict__ x, const float* __restrict__ ws,
    const float* __restrict__ bu1, const float* __restrict__ bu2,
    float* __restrict__ out) {
  __shared__ float lds_h[4 * 16 * LDS_STRIDE];
  const int lane = threadIdx.x & 31;
  const int wv   = threadIdx.x >> 5;
  const int tile = blockIdx.x * 4 + wv;
  if (tile * 16 >= N_NODES_C) return;
  const int base = tile * 16;
  const int half = lane >> 4;
  const int ln   = lane & 15;

  const float* px = x + (long long)(base + ln) * NODE_IN_C;
  const float* pa = out + (long long)(base + ln) * OUT_DIM_C;
  const float* Su1 = ws + OFF_SU1;
  const float* Su2 = ws + OFF_SU2;

  // ---- layer 1: h = relu([x | aggr] @ Wu1 + bu1), K = 256 ----
  v8f acc[NT];
#pragma unroll
  for (int nt = 0; nt < NT; ++nt) {
    float bv = bu1[nt * 16 + ln];
    acc[nt] = (v8f){bv, bv, bv, bv, bv, bv, bv, bv};
  }
  for (int kk = 0; kk < 32; ++kk) {                 // K 0..127 : x
    v2f a = *(const v2f*)(px + 4 * kk + 2 * half);
#pragma unroll
    for (int nt = 0; nt < NT; ++nt) {
      v2f b = *(const v2f*)(Su1 + ((nt * 64 + kk) * 32 + lane) * 2);
      acc[nt] = wmma_f32(a, b, acc[nt]);
    }
  }
  for (int kk = 32; kk < 64; ++kk) {                // K 128..255 : aggr
    v2f a = *(const v2f*)(pa + 4 * (kk - 32) + 2 * half);
#pragma unroll
    for (int nt = 0; nt < NT; ++nt) {
      v2f b = *(const v2f*)(Su1 + ((nt * 64 + kk) * 32 + lane) * 2);
      acc[nt] = wmma_f32(a, b, acc[nt]);
    }
  }

  float* hbuf = lds_h + wv * (16 * LDS_STRIDE);
#pragma unroll
  for (int nt = 0; nt < NT; ++nt)
#pragma unroll
    for (int j = 0; j < 8; ++j) {
      float v = acc[nt][j];
      hbuf[(j + 8 * half) * LDS_STRIDE + nt * 16 + ln] = v > 0.f ? v : 0.f;
    }

  // ---- layer 2: out = h @ Wu2 + bu2, K = 128 ----
  v8f acc2[NT];
#pragma unroll
  for (int nt = 0; nt < NT; ++nt) {
    float bv = bu2[nt * 16 + ln];
    acc2[nt] = (v8f){bv, bv, bv, bv, bv, bv, bv, bv};
  }
  for (int kk = 0; kk < 32; ++kk) {
    v2f a = *(const v2f*)(hbuf + ln * LDS_STRIDE + 4 * kk + 2 * half);
#pragma unroll
    for (int nt = 0; nt < NT; ++nt) {
      v2f b = *(const v2f*)(Su2 + ((nt * 32 + kk) * 32 + lane) * 2);
      acc2[nt] = wmma_f32(a, b, acc2[nt]);
    }
  }

  // store final rows (C layout: VGPR j -> row base + j + 8*half, col nt*16+ln)
#pragma unroll
  for (int j = 0; j < 8; ++j) {
    float* dst = out + (long long)(base + j + 8 * half) * OUT_DIM_C;
#pragma unroll
    for (int nt = 0; nt < NT; ++nt)
      dst[nt * 16 + ln] = acc2[nt][j];
  }
}

extern "C" void kernel_launch(void* const* d_in, const int* in_sizes, int n_in,
                              void* d_out, int out_size, void* d_ws, size_t ws_size,
                              hipStream_t stream) {
  (void)in_sizes; (void)n_in; (void)out_size; (void)ws_size;
  const float*     x   = (const float*)d_in[0];
  const long long* ei  = (const long long*)d_in[1];
  const float*     ea  = (const float*)d_in[2];
  const float*     Wm1 = (const float*)d_in[3];
  const float*     bm1 = (const float*)d_in[4];
  const float*     Wm2 = (const float*)d_in[5];
  const float*     bm2 = (const float*)d_in[6];
  const float*     Wu1 = (const float*)d_in[7];
  const float*     bu1 = (const float*)d_in[8];
  const float*     Wu2 = (const float*)d_in[9];
  const float*     bu2 = (const float*)d_in[10];
  float* out = (float*)d_out;
  float* ws  = (float*)d_ws;

  // d_out doubles as the aggregation buffer; zero it every call.
  int n4 = (N_NODES_C * OUT_DIM_C) / 4;
  zero_out_kernel<<<(n4 + 255) / 256, 256, 0, stream>>>((float4*)d_out, n4);
  prep_weights_kernel<<<(WS_TOTAL + 255) / 256, 256, 0, stream>>>(Wm1, Wm2, Wu1, Wu2, ws);

  int edge_blocks = (N_EDGES_C / 16 + 3) / 4;   // 37500 tiles -> 9375 blocks
  edge_mlp_kernel<<<edge_blocks, 128, 0, stream>>>(x, ei, ea, ws, bm1, bm2, out);

  int node_blocks = ((N_NODES_C + 15) / 16 + 3) / 4;  // 3125 tiles -> 782 blocks
  node_mlp_kernel<<<node_blocks, 128, 0, stream>>>(x, ws, bu1, bu2, out);
}